// EmpiricalCRPS_14645838479695
// MI455X (gfx1250) — compile-verified
//
#include <hip/hip_runtime.h>

typedef __attribute__((ext_vector_type(2))) float v2f;
typedef __attribute__((ext_vector_type(8))) float v8f;

#define M_ENS 16
#define N_TOT (8 * 512 * 512)   /* 2,097,152 (b,h,w) points            */
#define N_VEC (N_TOT / 4)       /* 524,288 float4 points               */
#define NBLK  1024
#define NTHR  256

// ---------------------------------------------------------------------------
// Pass 1: fused streaming reduction. Each thread handles float4 positions,
// loops all 16 ensemble members (16x global_load_b128, 8 MiB apart, fully
// unrolled for MLP latency hiding), accumulating both
//   accA = sum |x_i - y|           and
//   accS = sum (mean_i |x_i|)^2
// then block-reduces via LDS into per-block partials in d_ws.
// ---------------------------------------------------------------------------
__global__ __launch_bounds__(NTHR) void crps_partial_kernel(
    const float* __restrict__ preds, const float* __restrict__ yt,
    float* __restrict__ partials) {
  const float4* __restrict__ p4 = reinterpret_cast<const float4*>(preds);
  const float4* __restrict__ y4 = reinterpret_cast<const float4*>(yt);

  float accA = 0.0f;
  float accS = 0.0f;
  const int gstride = NBLK * NTHR;

  for (int idx = (int)blockIdx.x * NTHR + (int)threadIdx.x; idx < N_VEC;
       idx += gstride) {
    // CDNA5 speculative prefetch of next grid-stride iteration (OOB-safe).
    __builtin_prefetch(&y4[idx + gstride], 0, 0);
    __builtin_prefetch(&p4[idx + gstride], 0, 0);

    const float4 t = y4[idx];
    float ax = 0.f, ay = 0.f, az = 0.f, aw = 0.f;
    float sx = 0.f, sy = 0.f, sz = 0.f, sw = 0.f;
#pragma unroll
    for (int i = 0; i < M_ENS; ++i) {
      const float4 x = p4[(size_t)i * (size_t)N_VEC + (size_t)idx];
      ax += fabsf(x.x - t.x); sx += fabsf(x.x);
      ay += fabsf(x.y - t.y); sy += fabsf(x.y);
      az += fabsf(x.z - t.z); sz += fabsf(x.z);
      aw += fabsf(x.w - t.w); sw += fabsf(x.w);
    }
    accA += (ax + ay) + (az + aw);
    const float inv = 1.0f / (float)M_ENS;
    sx *= inv; sy *= inv; sz *= inv; sw *= inv;
    accS += (sx * sx + sy * sy) + (sz * sz + sw * sw);
  }

  __shared__ float red[NTHR];
  red[threadIdx.x] = accA;
  __syncthreads();
  for (int o = NTHR / 2; o > 0; o >>= 1) {
    if ((int)threadIdx.x < o) red[threadIdx.x] += red[threadIdx.x + o];
    __syncthreads();
  }
  if (threadIdx.x == 0) partials[blockIdx.x] = red[0];
  __syncthreads();

  red[threadIdx.x] = accS;
  __syncthreads();
  for (int o = NTHR / 2; o > 0; o >>= 1) {
    if ((int)threadIdx.x < o) red[threadIdx.x] += red[threadIdx.x + o];
    __syncthreads();
  }
  if (threadIdx.x == 0) partials[NBLK + blockIdx.x] = red[0];
}

// ---------------------------------------------------------------------------
// Pass 2: single wave32 (EXEC all ones, as WMMA requires). Lane-strided serial
// accumulation of the 1024 block partials, then a cross-lane sum via
// V_WMMA_F32_16X16X4_F32 with B = ones:
//   A layout (16x4 f32): lane L<16 -> A[L][0]=v[L];  lane L>=16 -> A[L-16][2]=v[L]
//   => D[m][n] = v[m] + v[m+16] for all n.
//   D layout: lane n<16 VGPR j holds D[j][n]; lane n>=16 holds D[j+8][n-16].
//   Per-lane sum of the 8 D regs gives S_low (lanes 0-15) / S_high (16-31);
//   lane 0 combines via LDS. Exact f32 math, fixed order -> deterministic.
// ---------------------------------------------------------------------------
__global__ __launch_bounds__(32) void crps_final_kernel(
    const float* __restrict__ partials, float* __restrict__ out) {
  const int lane = threadIdx.x;

  float vA = 0.0f, vS = 0.0f;
  for (int k = lane; k < NBLK; k += 32) {  // NBLK % 32 == 0: no divergence
    vA += partials[k];
    vS += partials[NBLK + k];
  }

  v2f aA = {vA, 0.0f};
  v2f aS = {vS, 0.0f};
  v2f b  = {1.0f, 1.0f};
  v8f z  = {};

  v8f dA = __builtin_amdgcn_wmma_f32_16x16x4_f32(
      /*neg_a=*/false, aA, /*neg_b=*/false, b,
      /*c_mod=*/(short)0, z, /*reuse_a=*/false, /*reuse_b=*/false);
  v8f dS = __builtin_amdgcn_wmma_f32_16x16x4_f32(
      /*neg_a=*/false, aS, /*neg_b=*/false, b,
      /*c_mod=*/(short)0, z, /*reuse_a=*/false, /*reuse_b=*/false);

  float sA = 0.0f, sS = 0.0f;
#pragma unroll
  for (int j = 0; j < 8; ++j) {
    sA += dA[j];
    sS += dS[j];
  }

  __shared__ float redA[32];
  __shared__ float redS[32];
  redA[lane] = sA;
  redS[lane] = sS;
  __syncthreads();

  if (lane == 0) {
    const float sumA = redA[0] + redA[16];  // total sum |x - y|
    const float sumS = redS[0] + redS[16];  // total sum s^2
    // first  = sumA / (M * N_TOT)           = sumA / 2^25
    // second = (sumS / N_TOT) / (2 * M * M) = sumS / 2^30
    const float first  = sumA * (1.0f / 33554432.0f);
    const float second = sumS * (1.0f / 1073741824.0f);
    out[0] = first - second;
  }
}

extern "C" void kernel_launch(void* const* d_in, const int* in_sizes, int n_in,
                              void* d_out, int out_size, void* d_ws,
                              size_t ws_size, hipStream_t stream) {
  (void)in_sizes; (void)n_in; (void)out_size; (void)ws_size;
  const float* preds = (const float*)d_in[0];  // (16, 8, 512, 512) f32
  const float* yt    = (const float*)d_in[1];  // (8, 512, 512) f32
  float* partials    = (float*)d_ws;           // 2 * NBLK floats (8 KiB)

  crps_partial_kernel<<<NBLK, NTHR, 0, stream>>>(preds, yt, partials);
  crps_final_kernel<<<1, 32, 0, stream>>>(partials, (float*)d_out);
}